// SubgraphGNNEncoder_16226386444324
// MI455X (gfx1250) — compile-verified
//
#include <hip/hip_runtime.h>

// Problem constants (match reference)
#define S_SUB 4096
#define KNODE 64
#define HDIM  128
#define NROWS (S_SUB * KNODE)
#define NGLOB 50000
#define NLAYER 5
#define LSTR  132          // LDS row stride (floats); 132 % 64 == 4 -> conflict-free frags
#define BN_EPS 1e-5f

typedef __attribute__((ext_vector_type(2))) float v2f;
typedef __attribute__((ext_vector_type(8))) float v8f;

__device__ __forceinline__ v8f wmma_k4(v2f a, v2f b, v8f c) {
  // D = A(16x4 f32) * B(4x16 f32) + C(16x16 f32)
  return __builtin_amdgcn_wmma_f32_16x16x4_f32(false, a, false, b, (short)0, c, false, false);
}

// C += A_lds[m0..m0+15][0..K) * W[0..K)[col-block], A in LDS (stride LSTR), W row-major ld=128
template<int K>
__device__ __forceinline__ v8f gemm_tile(const float* lA, int m0,
                                         const float* __restrict__ W, int col,
                                         int r0, int l16, v8f acc) {
#pragma unroll
  for (int kb = 0; kb < K; kb += 4) {
    const float* ap = lA + (m0 + l16) * LSTR + kb + r0;
    v2f a; a.x = ap[0]; a.y = ap[1];
    v2f b; b.x = W[(kb + r0) * HDIM + col];
    b.y = W[(kb + r0 + 1) * HDIM + col];
    acc = wmma_k4(a, b, acc);
  }
  return acc;
}

// ---------------- small precompute: collapse initializer into one 128x128 GEMM ----------------
__global__ __launch_bounds__(128) void k_precompute(
    const float* __restrict__ node_w, const float* __restrict__ node_b,
    const float* __restrict__ logp_w, const float* __restrict__ logp_b,
    const float* __restrict__ root_emb,
    const float* __restrict__ init_w, const float* __restrict__ init_b,
    float* __restrict__ W1c, float* __restrict__ v_lp,
    float* __restrict__ e0c, float* __restrict__ e1c, float* __restrict__ biasc) {
  int j = threadIdx.x;
  if (j >= HDIM) return;
  for (int i = 0; i < HDIM; ++i) {
    float s = 0.f;
    for (int k = 0; k < HDIM; ++k) s += node_w[i * HDIM + k] * init_w[k * HDIM + j];
    W1c[i * HDIM + j] = s;
  }
  float s1 = 0.f, sa = 0.f, sb = 0.f, sc = 0.f;
  for (int k = 0; k < HDIM; ++k) {
    float wb = init_w[(HDIM + k) * HDIM + j];
    float wc = init_w[(2 * HDIM + k) * HDIM + j];
    s1 += logp_w[k] * wb;
    sa += root_emb[k] * wc;
    sb += root_emb[HDIM + k] * wc;
    sc += node_b[k] * init_w[k * HDIM + j] + logp_b[k] * wb;
  }
  v_lp[j] = s1; e0c[j] = sa; e1c[j] = sb; biasc[j] = sc + init_b[j];
}

__global__ void k_zero(float* p, int n) {
  int i = blockIdx.x * blockDim.x + threadIdx.x;
  if (i < n) p[i] = 0.f;
}

__global__ void k_count(const int* __restrict__ nodes, float* __restrict__ cnt) {
  int i = blockIdx.x * 256 + threadIdx.x;
  if (i >= NROWS) return;
  int id = nodes[i];
  float vf = (id >= 0) ? 1.f : 0.f;
  int c = (id < 0) ? 0 : id;
  __hip_atomic_fetch_add(&cnt[c], vf, __ATOMIC_RELAXED, __HIP_MEMORY_SCOPE_AGENT);
}

// ---------------- initializer: h = vf*x@W1c + vf*lp*v_lp + root_emb_c + biasc ----------------
__global__ __launch_bounds__(256) void k_init(
    const float* __restrict__ x, const int* __restrict__ nodes,
    const int* __restrict__ target_nodes, const float* __restrict__ log_probs,
    const float* __restrict__ W1c, const float* __restrict__ v_lp,
    const float* __restrict__ e0c, const float* __restrict__ e1c,
    const float* __restrict__ biasc, float* __restrict__ h) {
  __shared__ float xt[KNODE * LSTR];
  __shared__ float vf_l[KNODE];
  __shared__ int rootidx;
  const int sub = blockIdx.x;
  const int tid = threadIdx.x;
  const int base = sub * KNODE;
  if (tid == 0) rootidx = KNODE;
  if (tid < KNODE) {
    int id = nodes[base + tid];
    vf_l[tid] = (id >= 0) ? 1.f : 0.f;
  }
  __syncthreads();
  if (tid < KNODE) {
    int rg = target_nodes[sub >> 2];
    if (nodes[base + tid] == rg) atomicMin(&rootidx, tid);
  }
  for (int i = tid; i < KNODE * HDIM; i += 256) {
    int r = i >> 7, c = i & (HDIM - 1);
    int id = nodes[base + r];
    int cl = (id < 0) ? 0 : id;
    xt[r * LSTR + c] = vf_l[r] * x[(size_t)cl * HDIM + c];
  }
  __syncthreads();
  int rl = (rootidx == KNODE) ? 0 : rootidx;  // argmax semantics (all-false -> 0)
  float lpv = log_probs[sub];
  lpv = __builtin_isfinite(lpv) ? lpv : 0.f;
  const int lane = tid & 31, wave = tid >> 5;
  const int half = lane >> 4, l16 = lane & 15;
  const int r0 = 2 * half;
  const int col = wave * 16 + l16;
  const float bias = biasc[col], vl = v_lp[col], e0 = e0c[col], e1 = e1c[col];
  for (int mt = 0; mt < 4; ++mt) {
    v8f acc;
#pragma unroll
    for (int r = 0; r < 8; ++r) acc[r] = bias;
    acc = gemm_tile<HDIM>(xt, mt * 16, W1c, col, r0, l16, acc);
#pragma unroll
    for (int r = 0; r < 8; ++r) {
      int row = mt * 16 + r + 8 * half;
      float v = acc[r] + vf_l[row] * lpv * vl + ((row == rl) ? e1 : e0);
      h[(size_t)(base + row) * HDIM + col] = v;
    }
  }
}

// ---------------- per-layer part 1: intra-subgraph GIN aggregation + 2-layer MLP ----------------
__global__ __launch_bounds__(256) void k_gin(
    const float* __restrict__ h, const int* __restrict__ nodes,
    const int* __restrict__ eidx, const int* __restrict__ eptr,
    const float* __restrict__ geps,
    const float* __restrict__ w1, const float* __restrict__ b1,
    const float* __restrict__ w2, const float* __restrict__ b2,
    float* __restrict__ gbuf, float* __restrict__ hsum, int E) {
  __shared__ float ht[KNODE * LSTR];  // h tile, then relu intermediate t
  __shared__ float gt[KNODE * LSTR];  // msg accumulator, then g
  __shared__ float vf_l[KNODE];
  __shared__ int cl_l[KNODE];
  const int sub = blockIdx.x;
  const int tid = threadIdx.x;
  const int base = sub * KNODE;
  if (tid < KNODE) {
    int id = nodes[base + tid];
    vf_l[tid] = (id >= 0) ? 1.f : 0.f;
    cl_l[tid] = (id < 0) ? 0 : id;
  }
  for (int i = tid; i < KNODE * HDIM; i += 256) {
    int r = i >> 7, c = i & (HDIM - 1);
    ht[r * LSTR + c] = h[(size_t)(base + r) * HDIM + c];
    gt[r * LSTR + c] = 0.f;
  }
  __syncthreads();
  // edge scatter-add in LDS (local src/dst in [0,64))
  int ea = eptr[sub], eb = eptr[sub + 1];
  int nE = eb - ea;
  for (int wi = tid; wi < nE * 32; wi += 256) {
    int e = ea + (wi >> 5);
    int f = (wi & 31) * 4;
    int sn = eidx[e];
    int dn = eidx[E + e];
#pragma unroll
    for (int q = 0; q < 4; ++q)
      atomicAdd(&gt[dn * LSTR + f + q], ht[sn * LSTR + f + q]);
  }
  __syncthreads();
  float ep = 1.f + geps[0];
  for (int i = tid; i < KNODE * HDIM; i += 256) {
    int r = i >> 7, c = i & (HDIM - 1);
    gt[r * LSTR + c] = ep * ht[r * LSTR + c] + gt[r * LSTR + c];
  }
  __syncthreads();
  const int lane = tid & 31, wave = tid >> 5;
  const int half = lane >> 4, l16 = lane & 15;
  const int r0 = 2 * half;
  const int col = wave * 16 + l16;
  const float bias1 = b1[col];
  for (int mt = 0; mt < 4; ++mt) {
    v8f acc;
#pragma unroll
    for (int r = 0; r < 8; ++r) acc[r] = bias1;
    acc = gemm_tile<HDIM>(gt, mt * 16, w1, col, r0, l16, acc);
#pragma unroll
    for (int r = 0; r < 8; ++r) {
      int row = mt * 16 + r + 8 * half;
      ht[row * LSTR + col] = fmaxf(acc[r], 0.f);  // t = relu(g@w1+b1), reuse ht
    }
  }
  __syncthreads();
  const float bias2 = b2[col];
  for (int mt = 0; mt < 4; ++mt) {
    v8f acc;
#pragma unroll
    for (int r = 0; r < 8; ++r) acc[r] = bias2;
    acc = gemm_tile<HDIM>(ht, mt * 16, w2, col, r0, l16, acc);
#pragma unroll
    for (int r = 0; r < 8; ++r) {
      int row = mt * 16 + r + 8 * half;
      float v = acc[r] * vf_l[row];
      gbuf[(size_t)(base + row) * HDIM + col] = v;
      __hip_atomic_fetch_add(&hsum[(size_t)cl_l[row] * HDIM + col], v,
                             __ATOMIC_RELAXED, __HIP_MEMORY_SCOPE_AGENT);
    }
  }
}

// ---------------- per-layer part 2: cross-subgraph gather + out proj + BN + residual ----------------
__global__ __launch_bounds__(256) void k_cross(
    const float* __restrict__ gbuf, const int* __restrict__ nodes,
    const float* __restrict__ hsum, const float* __restrict__ cnt,
    const float* __restrict__ ow, const float* __restrict__ ob,
    const float* __restrict__ gamma, const float* __restrict__ beta,
    float* __restrict__ h) {
  __shared__ float gt[KNODE * LSTR];
  __shared__ float vf_l[KNODE];
  __shared__ int cl_l[KNODE];
  __shared__ float sc_l[KNODE];
  const int sub = blockIdx.x;
  const int tid = threadIdx.x;
  const int base = sub * KNODE;
  if (tid < KNODE) {
    int id = nodes[base + tid];
    float vf = (id >= 0) ? 1.f : 0.f;
    int c = (id < 0) ? 0 : id;
    vf_l[tid] = vf;
    cl_l[tid] = c;
    sc_l[tid] = vf / fmaxf(cnt[c], 1.f);
  }
  for (int i = tid; i < KNODE * HDIM; i += 256) {
    int r = i >> 7, c = i & (HDIM - 1);
    gt[r * LSTR + c] = gbuf[(size_t)(base + r) * HDIM + c];
  }
  __syncthreads();
  const int lane = tid & 31, wave = tid >> 5;
  const int half = lane >> 4, l16 = lane & 15;
  const int r0 = 2 * half;
  const int col = wave * 16 + l16;
  const float bias = ob[col];
  const float gsc = gamma[col] * (1.0f / sqrtf(1.0f + BN_EPS));
  const float bt = beta[col];
  for (int mt = 0; mt < 4; ++mt) {
    v8f acc;
#pragma unroll
    for (int r = 0; r < 8; ++r) acc[r] = bias;
    // K = 0..127 : g part (LDS)
    acc = gemm_tile<HDIM>(gt, mt * 16, ow, col, r0, l16, acc);
    // K = 128..255 : x_cross part, A-fragments gathered straight from L2-resident hsum
    const int arow = mt * 16 + l16;
    const float* hs = hsum + (size_t)cl_l[arow] * HDIM;
    const float asc = sc_l[arow];
    const float* W2 = ow + HDIM * HDIM;
#pragma unroll
    for (int kb = 0; kb < HDIM; kb += 4) {
      v2f a; a.x = hs[kb + r0] * asc; a.y = hs[kb + r0 + 1] * asc;
      v2f b; b.x = W2[(kb + r0) * HDIM + col];
      b.y = W2[(kb + r0 + 1) * HDIM + col];
      acc = wmma_k4(a, b, acc);
    }
#pragma unroll
    for (int r = 0; r < 8; ++r) {
      int row = mt * 16 + r + 8 * half;
      size_t oi = (size_t)(base + row) * HDIM + col;
      float o = fmaxf(acc[r], 0.f) * gsc + bt;
      h[oi] = (o + h[oi]) * vf_l[row];
    }
  }
}

extern "C" void kernel_launch(void* const* d_in, const int* in_sizes, int n_in,
                              void* d_out, int out_size, void* d_ws, size_t ws_size,
                              hipStream_t stream) {
  const float* x = (const float*)d_in[0];
  const int* nodes = (const int*)d_in[1];
  const int* eidx = (const int*)d_in[2];
  const int* eptr = (const int*)d_in[3];
  const int* tgt = (const int*)d_in[4];
  const float* logp = (const float*)d_in[5];
  const float* npw = (const float*)d_in[6];
  const float* npb = (const float*)d_in[7];
  const float* lpw = (const float*)d_in[8];
  const float* lpb = (const float*)d_in[9];
  const float* remb = (const float*)d_in[10];
  const float* ipw = (const float*)d_in[11];
  const float* ipb = (const float*)d_in[12];
  const float* geps = (const float*)d_in[13];
  const float* gw1 = (const float*)d_in[14];
  const float* gb1 = (const float*)d_in[15];
  const float* gw2 = (const float*)d_in[16];
  const float* gb2 = (const float*)d_in[17];
  const float* ow = (const float*)d_in[18];
  const float* ob = (const float*)d_in[19];
  const float* gam = (const float*)d_in[20];
  const float* bet = (const float*)d_in[21];
  float* h = (float*)d_out;
  const int E = in_sizes[2] / 2;

  float* ws = (float*)d_ws;
  float* hsum = ws;                                // NGLOB*HDIM
  float* cnt = hsum + (size_t)NGLOB * HDIM;        // NGLOB (padded)
  float* W1c = cnt + 50048;                        // HDIM*HDIM
  float* vlp = W1c + HDIM * HDIM;
  float* e0c = vlp + HDIM;
  float* e1c = e0c + HDIM;
  float* bic = e1c + HDIM;
  float* gbuf = bic + HDIM;                        // NROWS*HDIM

  k_precompute<<<1, 128, 0, stream>>>(npw, npb, lpw, lpb, remb, ipw, ipb,
                                      W1c, vlp, e0c, e1c, bic);
  k_zero<<<(NGLOB + 255) / 256, 256, 0, stream>>>(cnt, NGLOB);
  k_count<<<(NROWS + 255) / 256, 256, 0, stream>>>(nodes, cnt);
  k_init<<<S_SUB, 256, 0, stream>>>(x, nodes, tgt, logp, W1c, vlp, e0c, e1c, bic, h);

  for (int l = 0; l < NLAYER; ++l) {
    k_zero<<<((NGLOB * HDIM) + 255) / 256, 256, 0, stream>>>(hsum, NGLOB * HDIM);
    k_gin<<<S_SUB, 256, 0, stream>>>(h, nodes, eidx, eptr, geps + l,
                                     gw1 + (size_t)l * HDIM * HDIM, gb1 + l * HDIM,
                                     gw2 + (size_t)l * HDIM * HDIM, gb2 + l * HDIM,
                                     gbuf, hsum, E);
    k_cross<<<S_SUB, 256, 0, stream>>>(gbuf, nodes, hsum, cnt,
                                       ow + (size_t)l * 2 * HDIM * HDIM, ob + l * HDIM,
                                       gam + l * HDIM, bet + l * HDIM, h);
  }
}